// GNNBase_87789131530402
// MI455X (gfx1250) — compile-verified
//
#include <hip/hip_runtime.h>
#include <hip/hip_bf16.h>

// ---------------------------------------------------------------------------
// GNNBase forward for MI455X (gfx1250, wave32, WMMA + async-to-LDS).
//  - critic branch reduced to the single requested node (node_index)
//  - actor branch = [B*N,128]x[128,256] fp32 GEMM via V_WMMA_F32_16X16X4_F32
//    with double-buffered GLOBAL_LOAD_ASYNC_TO_LDS_B128 tile staging
// ---------------------------------------------------------------------------

typedef __attribute__((ext_vector_type(2))) float v2f;
typedef __attribute__((ext_vector_type(8))) float v8f;

#define F_IN   128
#define HDIM   256
#define TNUM_OFFSET 8      // W_gcn rows 0..7 are the type one-hot rows
#define SEL_CAP 4096
#define NEG_SLOPE 0.01f

// ---------------- small helper kernels (critic preprocessing) --------------

__global__ void init_kernel(float* __restrict__ deg, int* __restrict__ selcnt, int N) {
    int i = blockIdx.x * blockDim.x + threadIdx.x;
    if (i < N) deg[i] = 1.0f;          // self-loop contributes 1 to every degree
    if (i == 0) *selcnt = 0;
}

__global__ void degree_kernel(const int* __restrict__ col, float* __restrict__ deg, int E) {
    int e = blockIdx.x * blockDim.x + threadIdx.x;
    if (e < E) atomicAdd(&deg[col[e]], 1.0f);
}

__global__ void dinv_kernel(const float* __restrict__ deg, float* __restrict__ dinv, int N) {
    int i = blockIdx.x * blockDim.x + threadIdx.x;
    if (i < N) {
        float d = deg[i];
        dinv[i] = (d > 0.0f) ? rsqrtf(d) : 0.0f;
    }
}

__global__ void select_kernel(const int* __restrict__ col, const int* __restrict__ node_idx,
                              int* __restrict__ sel, int* __restrict__ selcnt, int E) {
    int e = blockIdx.x * blockDim.x + threadIdx.x;
    if (e < E && col[e] == node_idx[0]) {
        int p = atomicAdd(selcnt, 1);
        if (p < SEL_CAP) sel[p] = e;
    }
}

// ------------- critic: value[b] for the single node of interest ------------
// grid = B blocks, 256 threads (thread = output channel h)
__global__ __launch_bounds__(256) void critic_kernel(
    const float* __restrict__ xs,        // [B, N, 128]
    const int*   __restrict__ erow,      // edge_index[0], source nodes
    const float* __restrict__ dinv,      // [N]
    const int*   __restrict__ sel,       // compacted edge ids with col==node
    const int*   __restrict__ selcnt,
    const float* __restrict__ Wg,        // [136, 256] row-major
    const float* __restrict__ bg,        // [256]
    const float* __restrict__ Wv,        // [256]
    const float* __restrict__ bv,        // [1]
    const int*   __restrict__ type_index,// [B]
    const int*   __restrict__ node_idx,  // [1]
    int N,
    float* __restrict__ out_value)       // [B]
{
    __shared__ float sh[F_IN];
    __shared__ float red[256];
    const int b = blockIdx.x;
    const int h = threadIdx.x;
    const int n100 = node_idx[0];
    const int t = type_index[b];
    const float wtype = Wg[t * HDIM + h];
    const float d100 = dinv[n100];
    const int nsel = min(selcnt[0], SEL_CAP);
    const size_t xbase = (size_t)b * N * F_IN;

    float acc = 0.0f;
    for (int s = 0; s <= nsel; ++s) {     // last iteration = self-loop of node
        int r; float norm;
        if (s < nsel) { int e = sel[s]; r = erow[e]; norm = dinv[r] * d100; }
        else          { r = n100;                    norm = d100 * d100;    }
        __syncthreads();
        if (h < F_IN) sh[h] = xs[xbase + (size_t)r * F_IN + h];
        __syncthreads();
        float dot = 0.0f;
        #pragma unroll 8
        for (int k = 0; k < F_IN; ++k)
            dot = fmaf(sh[k], Wg[(TNUM_OFFSET + k) * HDIM + h], dot);
        acc += norm * (dot + wtype);
    }
    float v = acc + bg[h];
    v = (v >= 0.0f) ? v : NEG_SLOPE * v;       // LeakyReLU
    red[h] = v * Wv[h];
    __syncthreads();
    for (int off = 128; off > 0; off >>= 1) {
        if (h < off) red[h] += red[h + off];
        __syncthreads();
    }
    if (h == 0) out_value[b] = red[0] + bv[0];
}

// --------------------- actor: fp32 WMMA GEMM + LeakyReLU -------------------
// out[m, h] = leaky( b[h] + cond[m]*W[0,h] + sum_k xs[m,k]*W[1+k,h] )
// block = 512 threads = 16 waves; wave w owns columns [16w, 16w+16).
// B fragments for the whole K=128 live in registers (loaded once per block).
// A tiles are double-buffered via GLOBAL_LOAD_ASYNC_TO_LDS_B128 (ASYNCcnt),
// overlapping the next tile's HBM fetch with the current tile's 32 WMMAs.
#define APAD 4
#define ASTRIDE (F_IN + APAD)   // 132 floats: conflict-free ds_load_b64 reads

__global__ __launch_bounds__(512) void actor_wmma_kernel(
    const float* __restrict__ xs,    // [M, 128]
    const float* __restrict__ cond,  // [M]
    const float* __restrict__ W,     // [129, 256] row-major (row 0 = cond)
    const float* __restrict__ bias,  // [256]
    float* __restrict__ out,         // [M, 256]
    int M)
{
    __shared__ float As[2][16 * ASTRIDE];   // 2 x 8448 B
    __shared__ float cs[2][16];

    const int tid  = threadIdx.x;
    const int lane = tid & 31;
    const int wave = tid >> 5;                // 0..15 -> column group
    const int col  = (wave << 4) + (lane & 15);
    const int half = lane >> 4;               // K sub-group per ISA A/B layout
    const int m    = lane & 15;

    // Per-thread slot in the cooperative tile load: thread i owns 16 bytes.
    const int lrow  = tid >> 5;               // (i*4)/128
    const int lcolk = (tid & 31) << 2;        // (i*4)%128
    // Flat pointers to LDS carry the byte offset in addr[31:0] (ISA 10.2).
    const unsigned lds0 = (unsigned)(uintptr_t)&As[0][lrow * ASTRIDE + lcolk];
    const unsigned lds1 = (unsigned)(uintptr_t)&As[1][lrow * ASTRIDE + lcolk];

    // Preload B fragments: step s covers K = 4s..4s+3 ; per-lane K = 4s+2*half+{0,1}
    v2f bfrag[32];
    #pragma unroll
    for (int s = 0; s < 32; ++s) {
        int k = 4 * s + 2 * half;
        bfrag[s].x = W[(size_t)(1 + k) * HDIM + col];
        bfrag[s].y = W[(size_t)(2 + k) * HDIM + col];
    }
    const float w0 = W[col];       // cond weight (row 0)
    const float bh = bias[col];

    const int mtiles = M >> 4;
    int t = blockIdx.x;

    // Prologue: kick off the async fill of buffer 0 with this block's first tile.
    if (t < mtiles) {
        unsigned long long ga =
            (unsigned long long)(uintptr_t)(xs + (size_t)t * (16 * F_IN)) + ((size_t)tid << 4);
        asm volatile("global_load_async_to_lds_b128 %0, %1, off"
                     :: "v"(lds0), "v"(ga) : "memory");
        if (tid < 16) cs[0][tid] = cond[t * 16 + tid];
    }

    int p = 0;
    for (; t < mtiles; t += gridDim.x, p ^= 1) {
        // Buffer p ready for every wave; all reads of buffer p^1 are done.
        asm volatile("s_wait_asynccnt 0x0" ::: "memory");
        __syncthreads();

        // Start the async fill of the other buffer with the next tile.
        int nt = t + gridDim.x;
        if (nt < mtiles) {
            unsigned long long ga =
                (unsigned long long)(uintptr_t)(xs + (size_t)nt * (16 * F_IN)) + ((size_t)tid << 4);
            unsigned dst = p ? lds0 : lds1;
            asm volatile("global_load_async_to_lds_b128 %0, %1, off"
                         :: "v"(dst), "v"(ga) : "memory");
            if (tid < 16) cs[p ^ 1][tid] = cond[nt * 16 + tid];
        }

        // C init: bias + cond * W[0]  (VGPR r holds row m = r + 8*half)
        const float* __restrict__ Ab = As[p];
        v8f c;
        #pragma unroll
        for (int r = 0; r < 8; ++r) c[r] = fmaf(cs[p][r + 8 * half], w0, bh);

        #pragma unroll
        for (int s = 0; s < 32; ++s) {
            v2f a = *(const v2f*)&Ab[m * ASTRIDE + 4 * s + 2 * half];
            c = __builtin_amdgcn_wmma_f32_16x16x4_f32(
                    false, a, false, bfrag[s], (short)0, c, false, false);
        }

        #pragma unroll
        for (int r = 0; r < 8; ++r) {
            float v = c[r];
            v = (v >= 0.0f) ? v : NEG_SLOPE * v;
            out[(size_t)(t * 16 + r + 8 * half) * HDIM + col] = v;
        }
    }
}

// scalar tail (only runs if M % 16 != 0; with these shapes it does not)
__global__ void actor_tail_kernel(const float* __restrict__ xs, const float* __restrict__ cond,
                                  const float* __restrict__ W, const float* __restrict__ bias,
                                  float* __restrict__ out, int m0, int M)
{
    int idx = blockIdx.x * blockDim.x + threadIdx.x;
    int row = m0 + idx / HDIM;
    int h = idx % HDIM;
    if (row >= M) return;
    float accv = fmaf(cond[row], W[h], bias[h]);
    for (int k = 0; k < F_IN; ++k)
        accv = fmaf(xs[(size_t)row * F_IN + k], W[(size_t)(1 + k) * HDIM + h], accv);
    out[(size_t)row * HDIM + h] = (accv >= 0.0f) ? accv : NEG_SLOPE * accv;
}

// ---------------------------------------------------------------------------

extern "C" void kernel_launch(void* const* d_in, const int* in_sizes, int n_in,
                              void* d_out, int out_size, void* d_ws, size_t ws_size,
                              hipStream_t stream) {
    const float* xs   = (const float*)d_in[0];   // [B,N,128]
    const int*   ei   = (const int*)  d_in[1];   // [2,E]
    const float* cond = (const float*)d_in[2];   // [B,N,1]
    const int*   tix  = (const int*)  d_in[3];   // [B]
    const int*   nidx = (const int*)  d_in[4];   // scalar
    const float* Wg   = (const float*)d_in[5];   // [136,256]
    const float* bg   = (const float*)d_in[6];   // [256]
    const float* Wa   = (const float*)d_in[7];   // [129,256]
    const float* ba   = (const float*)d_in[8];   // [256]
    const float* Wv   = (const float*)d_in[9];   // [256]
    const float* bv   = (const float*)d_in[10];  // [1]

    const int B  = in_sizes[3];
    const int BN = in_sizes[2];          // B*N (cond has 1 feature)
    const int N  = BN / B;
    const int E  = in_sizes[1] / 2;

    float* out_value = (float*)d_out;         // [B]
    float* out_actor = (float*)d_out + B;     // [B*N, 256]

    // workspace layout
    float* deg    = (float*)d_ws;             // [N]
    float* dinv   = deg + N;                  // [N]
    int*   sel    = (int*)(dinv + N);         // [SEL_CAP]
    int*   selcnt = sel + SEL_CAP;            // [1]

    const int T = 256;
    init_kernel  <<<(N + T - 1) / T, T, 0, stream>>>(deg, selcnt, N);
    degree_kernel<<<(E + T - 1) / T, T, 0, stream>>>(ei + E, deg, E);  // col = ei[1]
    dinv_kernel  <<<(N + T - 1) / T, T, 0, stream>>>(deg, dinv, N);
    select_kernel<<<(E + T - 1) / T, T, 0, stream>>>(ei + E, nidx, sel, selcnt, E);
    critic_kernel<<<B, 256, 0, stream>>>(xs, ei, dinv, sel, selcnt,
                                         Wg, bg, Wv, bv, tix, nidx, N, out_value);

    const int mtiles = BN >> 4;
    int grid = mtiles < 1250 ? (mtiles > 0 ? mtiles : 1) : 1250;  // 8 tiles/block
    if (mtiles > 0)
        actor_wmma_kernel<<<grid, 512, 0, stream>>>(xs, cond, Wa, ba, out_actor, BN);
    const int m0 = mtiles << 4;
    if (m0 < BN) {
        int rem = (BN - m0) * HDIM;
        actor_tail_kernel<<<(rem + T - 1) / T, T, 0, stream>>>(xs, cond, Wa, ba, out_actor, m0, BN);
    }
}